// ProtoNode_43894565765292
// MI455X (gfx1250) — compile-verified
//
#include <hip/hip_runtime.h>

// ---- problem dims ----
#define B_   2048
#define C_   64
#define W_   40
#define P_   8000
#define NCLS 200
#define NPAD 208            // 200 padded up to 13 tiles of 16

// ---- workspace layout (floats) ----
#define XW_OFF 0
#define XW_SZ  (W_ * B_ * C_)        // 5,242,880  (~21.0 MB)  xw[w][b][c]
#define PW_OFF (XW_OFF + XW_SZ)
#define PW_SZ  (W_ * NPAD * C_)      //   532,480  (~2.1 MB)   pw[w][kc][c]
#define WP_OFF (PW_OFF + PW_SZ)
#define WP_SZ  (NPAD * P_)           // 1,664,000  (~6.7 MB)   Wpad[kc][p]
// total ~29.8 MB of d_ws

typedef float v2f __attribute__((ext_vector_type(2)));
typedef float v8f __attribute__((ext_vector_type(8)));

__device__ __forceinline__ float wave_reduce_sum(float s) {
#pragma unroll
  for (int m = 16; m >= 1; m >>= 1) s += __shfl_xor(s, m, 32);
  return s;
}

// One wave per pw row (w, kc): normalize prototype p = w + 40*kc over C=64,
// store into pw[w*NPAD + kc][c]; zero the padding rows kc in [200,208).
__global__ void proto_norm_kernel(const float* __restrict__ proto,
                                  float* __restrict__ pw) {
  int wave = blockIdx.x * (blockDim.x >> 5) + (threadIdx.x >> 5);
  int lane = threadIdx.x & 31;
  int w  = wave / NPAD;
  int kc = wave % NPAD;
  float* dst = pw + (size_t)wave * C_;
  if (kc >= NCLS) { dst[lane] = 0.0f; dst[lane + 32] = 0.0f; return; }
  int p = w + W_ * kc;
  const float* src = proto + (size_t)p * C_;
  float v0 = src[lane];
  float v1 = src[lane + 32];
  float n = sqrtf(wave_reduce_sum(v0 * v0 + v1 * v1));
  float inv = 1.0f / fmaxf(n, 1e-12f);   // matches F.normalize eps semantics
  dst[lane]      = v0 * inv;
  dst[lane + 32] = v1 * inv;
}

// One wave per (b, w): L2-normalize conv_features over C, store transposed
// into xw[w][b][c] so the per-w GEMM reads contiguous K.
__global__ void feat_norm_kernel(const float* __restrict__ cf,
                                 float* __restrict__ xw) {
  int wave = blockIdx.x * (blockDim.x >> 5) + (threadIdx.x >> 5);
  int lane = threadIdx.x & 31;
  int b = wave / W_;
  int w = wave % W_;
  const float* src = cf + (size_t)b * (C_ * W_) + w;     // cf[b][c][0][w]
  float v0 = src[(size_t)lane * W_];
  float v1 = src[(size_t)(lane + 32) * W_];
  float n = sqrtf(wave_reduce_sum(v0 * v0 + v1 * v1));
  float inv = 1.0f / fmaxf(n, 1e-12f);
  float* dst = xw + ((size_t)w * B_ + b) * C_;
  dst[lane]      = v0 * inv;
  dst[lane + 32] = v1 * inv;
}

// Copy last_layer_w (200 x 8000) into Wpad (208 x 8000), zeroing pad rows.
__global__ void wpad_kernel(const float* __restrict__ Wm,
                            float* __restrict__ wp) {
  int idx = blockIdx.x * blockDim.x + threadIdx.x;
  if (idx >= NPAD * P_) return;
  int k = idx / P_;
  wp[idx] = (k < NCLS) ? Wm[idx] : 0.0f;
}

// sim GEMM: for each w, S_w = X_w (2048x64) * P_w^T (64x200).
// One wave per 32x16 output tile (two M-tiles share each B operand),
// V_WMMA_F32_16X16X4_F32, K=64 -> 32 wmma / 48 loads per wave.
// A layout (16x4 f32): lanes 0-15 carry M=lane, K={k,k+1}; lanes 16-31 K={k+2,k+3}.
// B layout (4x16 f32): symmetric with N on the lane index.
__global__ void sim_wmma_kernel(const float* __restrict__ xw,
                                const float* __restrict__ pw,
                                float* __restrict__ mind) {
  int wave = blockIdx.x * (blockDim.x >> 5) + (threadIdx.x >> 5);
  int lane = threadIdx.x & 31;
  int half = lane >> 4;
  int l16  = lane & 15;

  const int NT  = NPAD / 16;            // 13
  const int MT2 = B_ / 32;              // 64 double-M tiles
  const int TPW = MT2 * NT;             // 832 tiles per w
  int w   = wave / TPW;
  int rem = wave % TPW;
  int mt2 = rem / NT;
  int nt  = rem % NT;

  const float* A0 = xw + (size_t)w * B_ * C_   + (size_t)(mt2 * 32 + l16) * C_ + half * 2;
  const float* A1 = A0 + 16 * C_;
  const float* Bp = pw + (size_t)w * NPAD * C_ + (size_t)(nt * 16 + l16) * C_ + half * 2;

  v8f acc0 = {};
  v8f acc1 = {};
#pragma unroll
  for (int kk = 0; kk < C_ / 4; ++kk) {
    v2f b  = *(const v2f*)(Bp + kk * 4);
    v2f a0 = *(const v2f*)(A0 + kk * 4);
    v2f a1 = *(const v2f*)(A1 + kk * 4);
    acc0 = __builtin_amdgcn_wmma_f32_16x16x4_f32(false, a0, false, b,
                                                 (short)0, acc0, false, false);
    acc1 = __builtin_amdgcn_wmma_f32_16x16x4_f32(false, a1, false, b,
                                                 (short)0, acc1, false, false);
  }

  int kcls = nt * 16 + l16;
  if (kcls < NCLS) {
    int p = w + W_ * kcls;              // global prototype index
#pragma unroll
    for (int r = 0; r < 8; ++r) {       // D VGPR r: M = r + 8*half, N = l16
      int brow = mt2 * 32 + half * 8 + r;
      mind[(size_t)brow * P_ + p]        = -acc0[r];  // min_distances = -sim
      mind[(size_t)(brow + 16) * P_ + p] = -acc1[r];
    }
  }
}

// logits GEMM: logits[b,k] = sum_p sim[b,p]*W[k,p] = -(sum_p mind[b,p]*Wpad[k,p]).
// M=2048, N=208(pad), K=8000 -> 2x2000 f32 wmma steps per 32x16 tile.
__global__ void logits_wmma_kernel(const float* __restrict__ mind,
                                   const float* __restrict__ wp,
                                   float* __restrict__ logits) {
  int wave = blockIdx.x * (blockDim.x >> 5) + (threadIdx.x >> 5);
  int lane = threadIdx.x & 31;
  int half = lane >> 4;
  int l16  = lane & 15;
  const int NT = NPAD / 16;             // 13
  int mt2 = wave / NT;
  int nt  = wave % NT;

  const float* A0 = mind + (size_t)(mt2 * 32 + l16) * P_ + half * 2;
  const float* A1 = A0 + (size_t)16 * P_;
  const float* Bp = wp   + (size_t)(nt * 16 + l16) * P_ + half * 2;

  v8f acc0 = {};
  v8f acc1 = {};
#pragma unroll 4
  for (int kk = 0; kk < P_ / 4; ++kk) {
    v2f b  = *(const v2f*)(Bp + kk * 4);
    v2f a0 = *(const v2f*)(A0 + kk * 4);
    v2f a1 = *(const v2f*)(A1 + kk * 4);
    acc0 = __builtin_amdgcn_wmma_f32_16x16x4_f32(false, a0, false, b,
                                                 (short)0, acc0, false, false);
    acc1 = __builtin_amdgcn_wmma_f32_16x16x4_f32(false, a1, false, b,
                                                 (short)0, acc1, false, false);
  }

  int k = nt * 16 + l16;
  if (k < NCLS) {
#pragma unroll
    for (int r = 0; r < 8; ++r) {
      int brow = mt2 * 32 + half * 8 + r;
      logits[(size_t)brow * NCLS + k]        = -acc0[r];  // negate: sim = -mind
      logits[(size_t)(brow + 16) * NCLS + k] = -acc1[r];
    }
  }
}

extern "C" void kernel_launch(void* const* d_in, const int* in_sizes, int n_in,
                              void* d_out, int out_size, void* d_ws, size_t ws_size,
                              hipStream_t stream) {
  (void)in_sizes; (void)n_in; (void)out_size; (void)ws_size;
  const float* cf    = (const float*)d_in[0];   // (2048, 64, 1, 40) f32
  const float* proto = (const float*)d_in[1];   // (8000, 64, 1, 1)  f32
  const float* Wm    = (const float*)d_in[2];   // (200, 8000)       f32

  float* out    = (float*)d_out;
  float* logits = out;                          // (2048, 200)
  float* mind   = out + (size_t)B_ * NCLS;      // (2048, 8000)

  float* ws = (float*)d_ws;
  float* xw = ws + XW_OFF;
  float* pw = ws + PW_OFF;
  float* wp = ws + WP_OFF;

  // 8 waves (256 threads) per block; all grids divide evenly.
  proto_norm_kernel <<<(W_ * NPAD) / 8,             256, 0, stream>>>(proto, pw);
  feat_norm_kernel  <<<(B_ * W_) / 8,               256, 0, stream>>>(cf, xw);
  wpad_kernel       <<<(NPAD * P_ + 255) / 256,     256, 0, stream>>>(Wm, wp);
  sim_wmma_kernel   <<<(W_ * (B_/32) * (NPAD/16)) / 8, 256, 0, stream>>>(xw, pw, mind);
  logits_wmma_kernel<<<((B_/32) * (NPAD/16)) / 8,   256, 0, stream>>>(mind, wp, logits);
}